// WordEmbedding_63393717289434
// MI455X (gfx1250) — compile-verified
//
#include <hip/hip_runtime.h>

// WordEmbedding gather: out[t, :] = W[ids[t], :]
//   ids:  int32[8192]          (B=4, S=2048 flattened)
//   W:    float32[32000, 1024] (row 0 pre-zeroed = padding_idx)
//   out:  float32[8192, 1024]
//
// Pure bandwidth problem: ~64 MiB traffic -> ~2.8 us at 23.3 TB/s.
// One 256-thread block moves 8 rows: each thread one b128 per row,
// all 8 loads in flight before any store. Ids fetched as two uniform
// int4 loads (s_load_b128). Output uses non-temporal b128 stores so the
// 128 MB table stays resident in the 192 MB L2.

typedef __attribute__((ext_vector_type(4))) float v4f;
typedef __attribute__((ext_vector_type(4))) int   v4i;

constexpr int kDim         = 1024;  // floats per embedding row
constexpr int kDimShift    = 10;    // log2(kDim)
constexpr int kTokPerBlock = 8;     // rows per block (MLP depth)
constexpr int kThreads     = 256;   // 256 * 4 floats = 1024 = one row

__global__ __launch_bounds__(kThreads)
void embed_gather_kernel(const int* __restrict__ ids,
                         const float* __restrict__ w,
                         float* __restrict__ out,
                         int n_tokens) {
  const int t0  = blockIdx.x * kTokPerBlock;
  const int col = threadIdx.x << 2;          // float offset within row

  if (t0 + kTokPerBlock <= n_tokens) {
    // ---- fast path: full group of 8 tokens (uniform branch) ----
    // Two uniform 16B id fetches -> s_load_b128 each.
    const v4i* idv = (const v4i*)(ids + t0);
    v4i ra = idv[0];
    v4i rb = idv[1];
    int rows[kTokPerBlock] = {ra.x, ra.y, ra.z, ra.w, rb.x, rb.y, rb.z, rb.w};

    // 8 outstanding global_load_b128 per thread to cover HBM latency.
    v4f v[kTokPerBlock];
#pragma unroll
    for (int i = 0; i < kTokPerBlock; ++i) {
      const v4f* src =
          (const v4f*)(w + ((size_t)rows[i] << kDimShift) + col);
      v[i] = *src;
    }

    // Streaming NT b128 stores: don't evict the L2-resident table.
#pragma unroll
    for (int i = 0; i < kTokPerBlock; ++i) {
      v4f* dst = (v4f*)(out + ((size_t)(t0 + i) << kDimShift) + col);
      __builtin_nontemporal_store(v[i], dst);
    }
  } else {
    // ---- tail path: partial group (never taken for 8192 tokens) ----
#pragma unroll
    for (int i = 0; i < kTokPerBlock; ++i) {
      int t = t0 + i;
      if (t < n_tokens) {
        int row = ids[t];
        const v4f* src =
            (const v4f*)(w + ((size_t)row << kDimShift) + col);
        v4f val = *src;
        v4f* dst = (v4f*)(out + ((size_t)t << kDimShift) + col);
        __builtin_nontemporal_store(val, dst);
      }
    }
  }
}

extern "C" void kernel_launch(void* const* d_in, const int* in_sizes, int n_in,
                              void* d_out, int out_size, void* d_ws, size_t ws_size,
                              hipStream_t stream) {
  const int*   ids = (const int*)d_in[0];    // input_ids, int32
  const float* w   = (const float*)d_in[1];  // weight table, f32
  float*       out = (float*)d_out;          // f32 output

  const int n_tokens = in_sizes[0];          // 4 * 2048 = 8192
  const int blocks   = (n_tokens + kTokPerBlock - 1) / kTokPerBlock;

  embed_gather_kernel<<<blocks, kThreads, 0, stream>>>(ids, w, out, n_tokens);
}